// KGMPNNLayer_23854248362408
// MI455X (gfx1250) — compile-verified
//
#include <hip/hip_runtime.h>

typedef float v2f __attribute__((ext_vector_type(2)));
typedef float v8f __attribute__((ext_vector_type(8)));

#define IN_F   32
#define EF_F   16
#define KTOT   544      // 512 product rows + 32 bias rows
#define KSTEPS 136      // KTOT / 4
#define WAVES  8

// ---- order-preserving float<->uint encoding for atomic segment-max ----
__device__ __forceinline__ unsigned enc_f(float f) {
  unsigned u = __float_as_uint(f);
  return (u & 0x80000000u) ? ~u : (u | 0x80000000u);
}
__device__ __forceinline__ float dec_f(unsigned e) {
  return __uint_as_float((e & 0x80000000u) ? (e ^ 0x80000000u) : ~e);
}

// ---- K0: prepack W_cat[544,64] into per-lane WMMA-B fragment order ----
// float index = ((kb*32 + lane)*8) + nt*2 + j  -> two float4 loads per lane/kb
__global__ void prepack_B(const float* __restrict__ w0, const float* __restrict__ b0,
                          const float* __restrict__ w1, const float* __restrict__ b1,
                          float* __restrict__ Bp) {
  int idx = blockIdx.x * blockDim.x + threadIdx.x;
  if (idx >= KSTEPS * 32 * 8) return;
  int kb   = idx >> 8;
  int rem  = idx & 255;
  int lane = rem >> 3;
  int t    = rem & 7;
  int nt   = t >> 1;          // 0..3 : {rel0 cols0-15, rel0 cols16-31, rel1 lo, rel1 hi}
  int j    = t & 1;
  int r    = nt >> 1;
  int col  = ((nt & 1) << 4) + (lane & 15);
  int khalf = (lane >= 16) ? 2 : 0;   // B 4x16 f32 layout: lane-half selects K pair
  int k = 4 * kb + khalf + j;
  const float* w = r ? w1 : w0;
  const float* b = r ? b1 : b0;
  float v;
  if (k < 512) { int i = k >> 4, f = k & 15; v = w[f * (IN_F * IN_F) + i * IN_F + col]; }
  else         { int i = k - 512;            v = b[i * IN_F + col]; }
  Bp[idx] = v;
}

// ---- K1: out = bias broadcast; den = 0; smax = enc(-1e30) ----
__global__ void init_out(float* __restrict__ out, float* __restrict__ den,
                         unsigned* __restrict__ smax, const float* __restrict__ bias, int N) {
  int tid = blockIdx.x * blockDim.x + threadIdx.x;
  if (tid < N * 32) out[tid] = bias[tid & 31];
  if (tid < 2 * N) { den[tid] = 0.f; smax[tid] = enc_f(-1e30f); }
}

// ---- K2: per-edge attention score ----
__global__ void edge_score(const float* __restrict__ feat, const int* __restrict__ src,
                           const int* __restrict__ dst, const float* __restrict__ aw,
                           const float* __restrict__ ab, float* __restrict__ score, int E) {
  int e = blockIdx.x * blockDim.x + threadIdx.x;
  if (e >= E) return;
  const float* fs = feat + (size_t)src[e] * IN_F;
  const float* fd = feat + (size_t)dst[e] * IN_F;
  float s = ab[0];
#pragma unroll
  for (int i = 0; i < IN_F; ++i) s += fs[i] * aw[i] + fd[i] * aw[IN_F + i];
  score[e] = (s > 0.f) ? s : 0.01f * s;   // leaky_relu
}

// ---- K3: segment max per (relation, dst) ----
__global__ void seg_max(const float* __restrict__ score, const int* __restrict__ dst,
                        const int* __restrict__ etype, unsigned* __restrict__ smax,
                        int N, int E) {
  int e = blockIdx.x * blockDim.x + threadIdx.x;
  if (e >= E) return;
  atomicMax(&smax[etype[e] * N + dst[e]], enc_f(score[e]));
}

// ---- K4: ex = exp(s - smax), den += ex ----
__global__ void seg_exp(const float* __restrict__ score, const int* __restrict__ dst,
                        const int* __restrict__ etype, const unsigned* __restrict__ smax,
                        float* __restrict__ ex, float* __restrict__ den, int N, int E) {
  int e = blockIdx.x * blockDim.x + threadIdx.x;
  if (e >= E) return;
  int idx = etype[e] * N + dst[e];
  float v = expf(score[e] - dec_f(smax[idx]));
  ex[e] = v;
  atomicAdd(&den[idx], v);
}

// ---- K5: per-tile [16,544]x[544,64] WMMA GEMM + alpha-weighted scatter ----
__global__ void __launch_bounds__(32 * WAVES)
gemm_scatter(const float* __restrict__ feat, const float* __restrict__ efeat,
             const int* __restrict__ src, const int* __restrict__ dst,
             const int* __restrict__ etype, const float* __restrict__ ex,
             const float* __restrict__ den, const float* __restrict__ Bp,
             float* __restrict__ out, int N, int E) {
  __shared__ float zsh[WAVES][16 * 33];   // z_src rows, padded (conflict-free)
  __shared__ float esh[WAVES][16 * 17];   // efeat rows, padded
  __shared__ float ash[WAVES][16];        // alpha per edge
  __shared__ int   dsh[WAVES][16];        // dst per edge
  __shared__ int   rsh[WAVES][16];        // relation per edge
  __shared__ int   ssh[WAVES][16];        // src per edge

  int wave = threadIdx.x >> 5;
  int lane = threadIdx.x & 31;
  int tile = blockIdx.x * WAVES + wave;
  int e0 = tile * 16;
  if (e0 >= E) return;

  if (lane < 16) {
    int e = e0 + lane;
    bool ok = (e < E);
    int d = ok ? dst[e] : 0;
    int r = ok ? etype[e] : 0;
    float a = 0.f;
    if (ok) { float dn = den[r * N + d]; a = ex[e] / (dn > 0.f ? dn : 1.f); }
    ash[wave][lane] = a; dsh[wave][lane] = d; rsh[wave][lane] = r;
    ssh[wave][lane] = ok ? src[e] : -1;
  }
  asm volatile("s_wait_dscnt 0x0" ::: "memory");   // DS in-order per wave

  float* zp = zsh[wave];
  float* ep = esh[wave];
#pragma unroll 4
  for (int m = 0; m < 16; ++m) {
    int s = ssh[wave][m];
    zp[m * 33 + lane] = (s >= 0) ? feat[(size_t)s * IN_F + lane] : 0.f;
  }
#pragma unroll 4
  for (int m2 = 0; m2 < 8; ++m2) {
    int row = m2 * 2 + (lane >> 4);
    int e = e0 + row;
    ep[row * 17 + (lane & 15)] = (e < E) ? efeat[(size_t)e * EF_F + (lane & 15)] : 0.f;
  }
  asm volatile("s_wait_dscnt 0x0" ::: "memory");

  const float* zrow = zp + (lane & 15) * 33;   // A layout: M = lane mod 16
  const float* erow = ep + (lane & 15) * 17;
  int khalf = (lane >> 4) << 1;                // lane-half -> K {0,1} vs {2,3}
  const float4* bp4 = (const float4*)Bp;

  v8f acc0 = {}, acc1 = {}, acc2 = {}, acc3 = {};

  // product rows: k = i*16 + f, p = z_src[i]*efeat[f]
  for (int kb = 0; kb < 128; ++kb) {
    float4 wa = bp4[(size_t)(kb * 32 + lane) * 2];
    float4 wb = bp4[(size_t)(kb * 32 + lane) * 2 + 1];
    int k0 = 4 * kb + khalf;
    int i0 = k0 >> 4, f0 = k0 & 15;
    float zi = zrow[i0];
    v2f a; a.x = zi * erow[f0]; a.y = zi * erow[f0 + 1];
    v2f b0; b0.x = wa.x; b0.y = wa.y;
    v2f b1; b1.x = wa.z; b1.y = wa.w;
    v2f b2; b2.x = wb.x; b2.y = wb.y;
    v2f b3; b3.x = wb.z; b3.y = wb.w;
    acc0 = __builtin_amdgcn_wmma_f32_16x16x4_f32(false, a, false, b0, (short)0, acc0, false, false);
    acc1 = __builtin_amdgcn_wmma_f32_16x16x4_f32(false, a, false, b1, (short)0, acc1, false, false);
    acc2 = __builtin_amdgcn_wmma_f32_16x16x4_f32(false, a, false, b2, (short)0, acc2, false, false);
    acc3 = __builtin_amdgcn_wmma_f32_16x16x4_f32(false, a, false, b3, (short)0, acc3, false, false);
  }
  // bias rows: k = 512+i, p = z_src[i]
  for (int kb = 128; kb < KSTEPS; ++kb) {
    float4 wa = bp4[(size_t)(kb * 32 + lane) * 2];
    float4 wb = bp4[(size_t)(kb * 32 + lane) * 2 + 1];
    int i0 = 4 * kb + khalf - 512;
    v2f a; a.x = zrow[i0]; a.y = zrow[i0 + 1];
    v2f b0; b0.x = wa.x; b0.y = wa.y;
    v2f b1; b1.x = wa.z; b1.y = wa.w;
    v2f b2; b2.x = wb.x; b2.y = wb.y;
    v2f b3; b3.x = wb.z; b3.y = wb.w;
    acc0 = __builtin_amdgcn_wmma_f32_16x16x4_f32(false, a, false, b0, (short)0, acc0, false, false);
    acc1 = __builtin_amdgcn_wmma_f32_16x16x4_f32(false, a, false, b1, (short)0, acc1, false, false);
    acc2 = __builtin_amdgcn_wmma_f32_16x16x4_f32(false, a, false, b2, (short)0, acc2, false, false);
    acc3 = __builtin_amdgcn_wmma_f32_16x16x4_f32(false, a, false, b3, (short)0, acc3, false, false);
  }

  // C/D layout: acc[g] holds (M = g + 8*(lane>=16), N = lane&15)
  int mh = (lane >> 4) * 8;
  int colLo = lane & 15;
#pragma unroll
  for (int g = 0; g < 8; ++g) {
    int M = g + mh;
    int d = dsh[wave][M];
    int r = rsh[wave][M];
    float al = ash[wave][M];
    float vlo = (r == 0) ? acc0[g] : acc2[g];
    float vhi = (r == 0) ? acc1[g] : acc3[g];
    atomicAdd(&out[(size_t)d * 32 + colLo],      al * vlo);
    atomicAdd(&out[(size_t)d * 32 + 16 + colLo], al * vhi);
  }
}

extern "C" void kernel_launch(void* const* d_in, const int* in_sizes, int n_in,
                              void* d_out, int out_size, void* d_ws, size_t ws_size,
                              hipStream_t stream) {
  const float* feat   = (const float*)d_in[0];
  const float* efeat  = (const float*)d_in[1];
  const int*   src    = (const int*)d_in[2];
  const int*   dst    = (const int*)d_in[3];
  const int*   etype  = (const int*)d_in[4];
  const float* attn_w = (const float*)d_in[5];
  const float* attn_b = (const float*)d_in[6];
  const float* ef1_w  = (const float*)d_in[7];
  const float* ef1_b  = (const float*)d_in[8];
  const float* ef2_w  = (const float*)d_in[9];
  const float* ef2_b  = (const float*)d_in[10];
  const float* bias   = (const float*)d_in[11];
  int N = in_sizes[0] / IN_F;
  int E = in_sizes[2];
  float* out = (float*)d_out;

  // workspace layout (16B-aligned sections)
  char* ws = (char*)d_ws;
  float*    Bp    = (float*)ws;                                   // 34816 floats = 139264 B
  unsigned* smax  = (unsigned*)(ws + 139264);                     // 2N uints
  float*    den   = (float*)(ws + 139264 + (size_t)8 * N);        // 2N floats
  float*    score = (float*)(ws + 139264 + (size_t)16 * N);       // E floats
  float*    ex    = (float*)(ws + 139264 + (size_t)16 * N + (size_t)4 * E); // E floats

  prepack_B<<<(KSTEPS * 256 + 255) / 256, 256, 0, stream>>>(ef1_w, ef1_b, ef2_w, ef2_b, Bp);
  int initN = (N * 32 > 2 * N) ? N * 32 : 2 * N;
  init_out<<<(initN + 255) / 256, 256, 0, stream>>>(out, den, smax, bias, N);
  edge_score<<<(E + 255) / 256, 256, 0, stream>>>(feat, src, dst, attn_w, attn_b, score, E);
  seg_max<<<(E + 255) / 256, 256, 0, stream>>>(score, dst, etype, smax, N, E);
  seg_exp<<<(E + 255) / 256, 256, 0, stream>>>(score, dst, etype, smax, ex, den, N, E);
  int tiles = (E + 15) / 16;
  gemm_scatter<<<(tiles + WAVES - 1) / WAVES, 32 * WAVES, 0, stream>>>(
      feat, efeat, src, dst, etype, ex, den, Bp, out, N, E);
}